// DigitCapsLayer_20847771255215
// MI455X (gfx1250) — compile-verified
//
#include <hip/hip_runtime.h>
#include <hip/hip_bf16.h>

// CapsNet dynamic routing, fully fused for MI455X (gfx1250, wave32).
// u_hat recomputed from x,W (L2-resident) each pass via V_WMMA_F32_16X16X4_F32;
// logits never materialized (logits_t = u_hat . sum_{j<t} v_j, kept as vsum[B,C,D]).
// Round 4: keep base-pointer + immediate-offset addressing from round 3, but force
// unroll(1) on the p-loop — unroll 2 doubled the live u_hat set and caused scratch
// spills ("Folded Reload") in the hot loop under the 256-VGPR occupancy cap.

typedef __attribute__((ext_vector_type(2))) float v2f;
typedef __attribute__((ext_vector_type(4))) float v4f;
typedef __attribute__((ext_vector_type(8))) float v8f;

#define CB   512   // batch
#define CP   1152  // primary capsules
#define CC   10    // digit capsules
#define CD   16    // output capsule dim
#define CIN  8     // input capsule dim
#define PSPLIT 18  // p-chunks per b-tile
#define PCHUNK 64  // 1152 / 18
#define NWAVES 4   // waves per block (128 threads)
#define PPW    16  // p's per wave: 64 / 4

// LDS row stride (floats) per batch-row: 160 data + pad -> 164 (16B aligned;
// 164 mod 64 = 36 => 16 lanes hit 16 distinct 4-bank groups for b128 reads).
#define BBSTRIDE 164

static __device__ __forceinline__ v8f v8f_zero() {
  v8f z;
#pragma unroll
  for (int i = 0; i < 8; ++i) z[i] = 0.0f;
  return z;
}

// One routing pass: for each (b,p,c) recompute u_hat tile with WMMA,
// logits = u_hat . vsum, coup = softmax_c(logits), s += coup * u_hat.
// vsum == zeros gives uniform coupling (iteration 0), matching softmax(0).
__global__ __launch_bounds__(NWAVES * 32, 4)
void caps_route_pass(const float* __restrict__ x,      // [B,P,IN]
                     const float* __restrict__ W,      // [P,C,D,IN]
                     const float* __restrict__ vsum,   // [B,C,D] running sum of v_j
                     float* __restrict__ s)            // [B,C,D] accumulator (pre-zeroed)
{
  __shared__ __align__(16) float lds_vs[16 * BBSTRIDE];            // vsum tile
  __shared__ __align__(16) float lds_acc[NWAVES * 16 * BBSTRIDE];  // per-wave s partials

  const int tid  = threadIdx.x;
  const int lane = tid & 31;
  const int wave = tid >> 5;
  const int bb   = lane & 15;   // column (batch) within tile / row (d) for A loads
  const int hi   = lane >> 4;   // lane-half selects K pair and d-half of C/D tile
  const int btile = blockIdx.x * 16;
  const int b     = btile + bb;

  // Stage vsum[btile:btile+16, :, :] (2560 floats) into LDS, padded rows.
#pragma unroll
  for (int k = 0; k < 20; ++k) {
    const int o   = tid + k * 128;          // 0..2559
    const int obb = o / (CC * CD);
    const int rem = o - obb * (CC * CD);
    lds_vs[obb * BBSTRIDE + rem] = vsum[(btile + obb) * (CC * CD) + rem];
  }
  __syncthreads();

  v8f sacc[CC];
#pragma unroll
  for (int c = 0; c < CC; ++c) sacc[c] = v8f_zero();

  const int pbase  = blockIdx.y * PCHUNK + wave;
  const int vsbase = bb * BBSTRIDE + 8 * hi;

  // Running base pointers; all per-capsule loads use constant immediate offsets.
  const float* wptr = W + (size_t)pbase * (CC * CD * CIN) + (size_t)(bb * CIN + 2 * hi);
  const float* xptr = x + (size_t)b * (CP * CIN) + (size_t)(pbase * CIN + 2 * hi);

#pragma unroll 1
  for (int j = 0; j < PPW; ++j) {
    // B-matrix (4i x 16b) from x[b,p,:]: lane half selects i-pair {2*hi, 2*hi+1} (+4)
    const v2f Blo = *(const v2f*)xptr;        // i = 2*hi, 2*hi+1
    const v2f Bhi = *(const v2f*)(xptr + 4);  // i = 2*hi+4, 2*hi+5

    v8f   uh[CC];
    float lg[CC];

#pragma unroll
    for (int c = 0; c < CC; ++c) {
      // A-matrix (16d x 4i) from W[p,c,:,:]: row d = bb, K pair from lane half.
      // Offset c*128 floats = 512c bytes -> folds into the load's 24-bit IOFFSET.
      const float* wp = wptr + c * (CD * CIN);
      const v2f Alo = *(const v2f*)wp;
      const v2f Ahi = *(const v2f*)(wp + 4);

      v8f acc = v8f_zero();
      acc = __builtin_amdgcn_wmma_f32_16x16x4_f32(false, Alo, false, Blo,
                                                  (short)0, acc, false, false);
      acc = __builtin_amdgcn_wmma_f32_16x16x4_f32(false, Ahi, false, Bhi,
                                                  (short)0, acc, false, false);
      uh[c] = acc;  // lane: b = bb, rows r -> d = r + 8*hi

      // agreement over d: vs from LDS (loop-invariant, compiler hoists/caches),
      // 8 in-lane FMAs + one cross-half xor-16 shuffle
      const float* vsp = &lds_vs[vsbase + c * CD];
      const v4f v0 = *(const v4f*)vsp;
      const v4f v1 = *(const v4f*)(vsp + 4);
      float t = 0.0f;
#pragma unroll
      for (int r = 0; r < 4; ++r) t = fmaf(acc[r], v0[r], t);
#pragma unroll
      for (int r = 0; r < 4; ++r) t = fmaf(acc[r + 4], v1[r], t);
      t += __shfl_xor(t, 16, 32);
      lg[c] = t;
    }

    // softmax over c (all 10 values per-lane, in registers)
    float m = lg[0];
#pragma unroll
    for (int c = 1; c < CC; ++c) m = fmaxf(m, lg[c]);
    float sum = 0.0f;
#pragma unroll
    for (int c = 0; c < CC; ++c) { lg[c] = __expf(lg[c] - m); sum += lg[c]; }
    const float inv = 1.0f / sum;

#pragma unroll
    for (int c = 0; c < CC; ++c) {
      const float coup = lg[c] * inv;
#pragma unroll
      for (int r = 0; r < 8; ++r) sacc[c][r] = fmaf(coup, uh[c][r], sacc[c][r]);
    }

    wptr += NWAVES * (CC * CD * CIN);  // next p for this wave (stride 4 p's)
    xptr += NWAVES * CIN;
  }

  // Per-wave partial s -> private LDS slice (plain stores, no atomics).
#pragma unroll
  for (int c = 0; c < CC; ++c) {
    float* ap = &lds_acc[(wave * 16 + bb) * BBSTRIDE + c * CD + 8 * hi];
    v4f lo, hh;
#pragma unroll
    for (int r = 0; r < 4; ++r) { lo[r] = sacc[c][r]; hh[r] = sacc[c][r + 4]; }
    *(v4f*)ap = lo;
    *(v4f*)(ap + 4) = hh;
  }
  __syncthreads();

  // Cross-wave reduce in LDS, then one global atomicAdd per output element.
#pragma unroll
  for (int k = 0; k < 20; ++k) {
    const int o   = tid + k * 128;          // 0..2559
    const int obb = o / (CC * CD);
    const int rem = o - obb * (CC * CD);
    float acc = 0.0f;
#pragma unroll
    for (int w = 0; w < NWAVES; ++w)
      acc += lds_acc[(w * 16 + obb) * BBSTRIDE + rem];
    atomicAdd(&s[(btile + obb) * (CC * CD) + rem], acc);
  }
}

// v = squash(s); vsum += v; optionally emit v; re-zero s for the next pass.
__global__ void caps_squash(float* __restrict__ s, float* __restrict__ vsum,
                            float* __restrict__ out)
{
  const int idx = blockIdx.x * blockDim.x + threadIdx.x;  // (b,c)
  if (idx >= CB * CC) return;
  float* sp = s + idx * CD;
  float vals[CD];
  float sq = 0.0f;
#pragma unroll
  for (int k = 0; k < 4; ++k) {
    v4f t = *(const v4f*)(sp + 4 * k);
#pragma unroll
    for (int e = 0; e < 4; ++e) { vals[4 * k + e] = t[e]; sq = fmaf(t[e], t[e], sq); }
  }
  const float scale = (sq / (1.0f + sq)) * rsqrtf(sq + 1e-7f);
  float* vp = vsum + idx * CD;
#pragma unroll
  for (int d = 0; d < CD; ++d) {
    const float v = vals[d] * scale;
    vp[d] += v;
    if (out) out[idx * CD + d] = v;
    sp[d] = 0.0f;  // re-zero s accumulator for next pass
  }
}

__global__ void caps_zero(float* __restrict__ p, int n)
{
  const int i = blockIdx.x * blockDim.x + threadIdx.x;
  if (i < n) p[i] = 0.0f;
}

extern "C" void kernel_launch(void* const* d_in, const int* in_sizes, int n_in,
                              void* d_out, int out_size, void* d_ws, size_t ws_size,
                              hipStream_t stream)
{
  const float* x = (const float*)d_in[0];   // [B,P,IN]
  const float* W = (const float*)d_in[1];   // [P,C,D,IN]
  float* out = (float*)d_out;               // [B,1,C,D,1] -> flat B*C*D

  float* vsum = (float*)d_ws;               // B*C*D floats
  float* s    = vsum + CB * CC * CD;        // B*C*D floats

  const int nstate = 2 * CB * CC * CD;      // 163,840 floats (~640 KB)
  caps_zero<<<(nstate + 255) / 256, 256, 0, stream>>>(vsum, nstate);

  const dim3 pgrid(CB / 16, PSPLIT, 1);     // 32 x 18 = 576 blocks
  const dim3 pblk(NWAVES * 32, 1, 1);       // 128 threads = 4 wave32
  const int sq_grid = (CB * CC + 255) / 256;

  // iter 0: vsum==0 -> softmax(0) = uniform 1/C
  caps_route_pass<<<pgrid, pblk, 0, stream>>>(x, W, vsum, s);
  caps_squash<<<sq_grid, 256, 0, stream>>>(s, vsum, nullptr);      // vsum = v0
  // iter 1: logits = u_hat . v0
  caps_route_pass<<<pgrid, pblk, 0, stream>>>(x, W, vsum, s);
  caps_squash<<<sq_grid, 256, 0, stream>>>(s, vsum, nullptr);      // vsum = v0+v1
  // iter 2: logits = u_hat . (v0+v1)
  caps_route_pass<<<pgrid, pblk, 0, stream>>>(x, W, vsum, s);
  caps_squash<<<sq_grid, 256, 0, stream>>>(s, vsum, out);          // final v -> d_out
}